// GaussianSplatRasterizerAntialiased_49675591745760
// MI455X (gfx1250) — compile-verified
//
#include <hip/hip_runtime.h>

typedef __attribute__((ext_vector_type(2))) float v2f;
typedef __attribute__((ext_vector_type(8))) float v8f;

#define NPIX 512
#define NVCH 64
#define CUBE_ELEMS (NVCH * NPIX * NPIX)   // 16,777,216 floats = 64 MB

__constant__ float kPIXSCALE = 0.05f;

__device__ __forceinline__ int refl(int p) {
    // jnp.pad(..., mode="reflect"): -1 -> 1, N -> N-2
    p = (p < 0) ? -p : p;
    p = (p > NPIX - 1) ? (2 * (NPIX - 1) - p) : p;
    return p;
}

__device__ __forceinline__ float gsel(const float* g, int d) {
    // g[d] for d in [0,6], else 0; branch-free selects (d is per-lane)
    float r = 0.0f;
    r = (d == 0) ? g[0] : r;
    r = (d == 1) ? g[1] : r;
    r = (d == 2) ? g[2] : r;
    r = (d == 3) ? g[3] : r;
    r = (d == 4) ? g[4] : r;
    r = (d == 5) ? g[5] : r;
    r = (d == 6) ? g[6] : r;
    return r;
}

// ---------------------------------------------------------------- zero cube
__global__ void zero_cube_kernel(float4* __restrict__ p, int n4) {
    int i = blockIdx.x * blockDim.x + threadIdx.x;
    int stride = gridDim.x * blockDim.x;
    float4 z = make_float4(0.f, 0.f, 0.f, 0.f);
    for (; i < n4; i += stride) p[i] = z;
}

// ---------------------------------------------------------------- splat
// Trilinear scatter-add into the (NV, NPIX, NPIX) cube. Cube (64 MB) is
// L2-resident (192 MB L2), so native global_atomic_add_f32 is the fast path.
__global__ void splat_kernel(const float2* __restrict__ pos,
                             const float* __restrict__ vel,
                             const float* __restrict__ flx,
                             float* __restrict__ cube, int M) {
    int i = blockIdx.x * blockDim.x + threadIdx.x;
    if (i >= M) return;

    const float FOV_HALF = 0.5f * (NPIX - 1) * 0.05f;  // 12.775
    const float VEL0 = -400.0f;
    const float DVI = 1.0f / 12.5f;

    float2 p = pos[i];
    float x = (p.x + FOV_HALF) / kPIXSCALE;
    float y = (p.y + FOV_HALF) / kPIXSCALE;
    float v = (vel[i] - VEL0) * DVI;

    float fx0 = floorf(x), fy0 = floorf(y), fv0 = floorf(v);
    int ix0 = (int)fx0, iy0 = (int)fy0, iv0 = (int)fv0;
    float fx = x - fx0, fy = y - fy0, fv = v - fv0;

    bool ok = (ix0 >= 0) & (ix0 < NPIX - 1) &
              (iy0 >= 0) & (iy0 < NPIX - 1) &
              (iv0 >= 0) & (iv0 < NVCH - 1);
    float f = ok ? flx[i] : 0.0f;

    ix0 = min(max(ix0, 0), NPIX - 2);
    iy0 = min(max(iy0, 0), NPIX - 2);
    iv0 = min(max(iv0, 0), NVCH - 2);

    float wx0 = 1.f - fx, wy0 = 1.f - fy, wv0 = 1.f - fv;

    size_t base = ((size_t)iv0 * NPIX + iy0) * NPIX + ix0;
    const size_t P = NPIX, PP = (size_t)NPIX * NPIX;
    unsafeAtomicAdd(cube + base,              f * wx0 * wy0 * wv0);
    unsafeAtomicAdd(cube + base + P,          f * wx0 * fy  * wv0);
    unsafeAtomicAdd(cube + base + 1,          f * fx  * wy0 * wv0);
    unsafeAtomicAdd(cube + base + P + 1,      f * fx  * fy  * wv0);
    unsafeAtomicAdd(cube + base + PP,         f * wx0 * wy0 * fv);
    unsafeAtomicAdd(cube + base + PP + P,     f * wx0 * fy  * fv);
    unsafeAtomicAdd(cube + base + PP + 1,     f * fx  * wy0 * fv);
    unsafeAtomicAdd(cube + base + PP + P + 1, f * fx  * fy  * fv);
}

// ---------------------------------------------------------------- blur (WMMA)
// Separable 7x7 Gaussian as banded matmuls on the matrix cores.
// Per wave32: one 16x16 output tile.
//   Horizontal: H(32x16) = In(32x24) x G(24x16),  G[k][n] = g1[k-n]
//   Vertical:   Out(16x16) = Gt(16x24) x H(24x16), Gt[m][k] = g1[k-m]
// K padded 22->24 (zero-weight bands), 6 k-steps of v_wmma_f32_16x16x4_f32
// per matmul => 18 WMMAs/tile. G (B-layout) and Gt (A-layout) share the same
// per-lane registers, so GB[] serves both operands.
#define WAVES_PER_BLOCK 8
#define BLUR_BLOCKS 2048

__global__ void __launch_bounds__(32 * WAVES_PER_BLOCK)
blur_wmma_kernel(const float* __restrict__ cube,
                 const float* __restrict__ k2d,
                 float* __restrict__ out) {
    __shared__ float Hlds[WAVES_PER_BLOCK][32 * 16];

    const int lane = threadIdx.x & 31;
    const int warp = threadIdx.x >> 5;
    const int ml = lane & 15;    // M (A) or N (B/C/D) index for this lane
    const int hi = lane >> 4;    // lane-half selector
    float* H = &Hlds[warp][0];

    // Recover the 1D gaussian: kernel2d = g1 outer g1, sum(g1)=1 => row sums.
    float g[7];
#pragma unroll
    for (int r = 0; r < 7; ++r) {
        float s = 0.f;
#pragma unroll
        for (int c = 0; c < 7; ++c) s += k2d[r * 7 + c];
        g[r] = s;
    }

    // Banded-matrix operand registers, per k-step kk (K = 4*kk + 2*hi + {0,1}).
    v2f GB[6];
#pragma unroll
    for (int kk = 0; kk < 6; ++kk) {
        int k0 = 4 * kk + 2 * hi;
        GB[kk].x = gsel(g, k0 - ml);
        GB[kk].y = gsel(g, k0 + 1 - ml);
    }

    const int NT = NVCH * (NPIX / 16) * (NPIX / 16);  // 65536 tiles
    const int nwaves = gridDim.x * WAVES_PER_BLOCK;   // wave-uniform loop

    for (int t = blockIdx.x * WAVES_PER_BLOCK + warp; t < NT; t += nwaves) {
        const int tx = t & 31;
        const int ty = (t >> 5) & 31;
        const int tv = t >> 10;
        const int x0 = tx * 16, y0 = ty * 16;
        const float* cp = cube + (size_t)tv * NPIX * NPIX;

        // ---- horizontal pass: rows [y0-3, y0+29) in two 16-row groups ----
        v8f Hlo = {}; v8f Hhi = {};
#pragma unroll
        for (int kk = 0; kk < 6; ++kk) {
            int c0 = 4 * kk + 2 * hi;
            int cx0 = refl(x0 - 3 + c0);
            int cx1 = refl(x0 - 3 + c0 + 1);
            int ry0 = refl(y0 - 3 + ml);
            int ry1 = refl(y0 + 13 + ml);
            v2f a0 = { cp[ry0 * NPIX + cx0], cp[ry0 * NPIX + cx1] };
            v2f a1 = { cp[ry1 * NPIX + cx0], cp[ry1 * NPIX + cx1] };
            Hlo = __builtin_amdgcn_wmma_f32_16x16x4_f32(
                false, a0, false, GB[kk], (short)0, Hlo, false, false);
            Hhi = __builtin_amdgcn_wmma_f32_16x16x4_f32(
                false, a1, false, GB[kk], (short)0, Hhi, false, false);
        }

        // ---- restage H from C/D layout into row-major LDS (32 rows x 16) ----
#pragma unroll
        for (int vi = 0; vi < 8; ++vi) {
            H[(vi + 8 * hi) * 16 + ml]      = Hlo[vi];
            H[(16 + vi + 8 * hi) * 16 + ml] = Hhi[vi];
        }

        // ---- vertical pass: Out = Gt x H (B operand streamed from LDS) ----
        v8f Out = {};
#pragma unroll
        for (int kk = 0; kk < 6; ++kk) {
            int k0 = 4 * kk + 2 * hi;
            v2f b = { H[k0 * 16 + ml], H[(k0 + 1) * 16 + ml] };
            Out = __builtin_amdgcn_wmma_f32_16x16x4_f32(
                false, GB[kk], false, b, (short)0, Out, false, false);
        }

        // ---- store 16x16 tile (D layout: VGPR vi holds rows vi, vi+8) ----
        float* op = out + ((size_t)tv * NPIX + y0) * NPIX + x0;
#pragma unroll
        for (int vi = 0; vi < 8; ++vi) {
            op[(vi + 8 * hi) * NPIX + ml] = Out[vi];
        }
    }
}

// ---------------------------------------------------------------- launch
extern "C" void kernel_launch(void* const* d_in, const int* in_sizes, int n_in,
                              void* d_out, int out_size, void* d_ws, size_t ws_size,
                              hipStream_t stream) {
    const float* pos = (const float*)d_in[0];   // (M,2) f32
    const float* vel = (const float*)d_in[1];   // (M,)  f32
    const float* flx = (const float*)d_in[2];   // (M,)  f32
    const float* k2d = (const float*)d_in[3];   // (7,7) f32
    float* outp = (float*)d_out;                // (64,512,512) f32
    float* cube = (float*)d_ws;                 // 64 MB scratch cube

    const int M = in_sizes[1];

    // 1) zero the accumulation cube
    zero_cube_kernel<<<4096, 256, 0, stream>>>((float4*)cube, CUBE_ELEMS / 4);

    // 2) trilinear scatter-add
    splat_kernel<<<(M + 255) / 256, 256, 0, stream>>>(
        (const float2*)pos, vel, flx, cube, M);

    // 3) 7x7 gaussian blur on the matrix cores
    blur_wmma_kernel<<<BLUR_BLOCKS, 32 * WAVES_PER_BLOCK, 0, stream>>>(
        cube, k2d, outp);
}